// EdgeCNNClassify_5214090297367
// MI455X (gfx1250) — compile-verified
//
#include <hip/hip_runtime.h>

// EdgeCNN pairwise classifier, algebraically folded:
//   out[b,i,j,o] = x[b,min(i,j)] . Wlo[o] + x[b,max(i,j)] . Whi[o] + obias[o]
// where Wlo = Wc[:, :100] @ W1, Whi = Wc[:, 100:] @ W1  (precomputed into d_ws),
// obias[o] = b1 . (Wc[o,:100] + Wc[o,100:]) + bc[o].
// Main kernel: one wave32 per batch element, GEMM via V_WMMA_F32_16X16X4_F32,
// M=16 (13 atoms padded), N=16 (8 used), K=512 (128 WMMAs). Folded weights are
// staged global->LDS with the Tensor Data Mover (tensor_load_to_lds +
// s_wait_tensorcnt), then a per-wave combine emits the [13,13,4] block.

typedef float v2f __attribute__((ext_vector_type(2)));
typedef float v8f __attribute__((ext_vector_type(8)));
typedef unsigned int u32x4 __attribute__((ext_vector_type(4)));
typedef int i32x4 __attribute__((ext_vector_type(4)));
typedef int i32x8 __attribute__((ext_vector_type(8)));

#define DM 512          // d_model
#define NATOMS 13
#define OUT_PER_B (NATOMS * NATOMS * 4)   // 676

// ---------------- pre-kernel: fold Wc into W1, build obias -------------------
// ws float layout:
//   [0, 8192)   : Wfold, "K-pair" layout: element (k,n) at ((k>>1)*16 + n)*2 + (k&1)
//                 n=0..3 -> Wlo rows, n=4..7 -> Whi rows, n=8..15 -> zero pad
//   [8192,8196) : obias[4]
__global__ __launch_bounds__(256) void edgecnn_fold_kernel(
    const float* __restrict__ W1,   // [100][512]
    const float* __restrict__ b1,   // [100]
    const float* __restrict__ Wc,   // [4][200]
    const float* __restrict__ bc,   // [4]
    float* __restrict__ ws) {
  int t = blockIdx.x * 256 + threadIdx.x;   // 0..8191
  int k = t >> 4;                           // 0..511  (d_model index)
  int n = t & 15;                           // 0..15   (padded output col)
  float acc = 0.0f;
  if (n < 8) {
    const float* wrow = Wc + (n & 3) * 200 + ((n < 4) ? 0 : 100);
    #pragma unroll 4
    for (int c = 0; c < 100; ++c) acc += wrow[c] * W1[c * DM + k];
  }
  ws[(((k >> 1) * 16 + n) << 1) + (k & 1)] = acc;
  if (t < 4) {
    float s = bc[t];
    for (int c = 0; c < 100; ++c)
      s += b1[c] * (Wc[t * 200 + c] + Wc[t * 200 + 100 + c]);
    ws[8192 + t] = s;
  }
}

// ---------------- main kernel: TDM stage + WMMA GEMM + combine ---------------
__global__ __launch_bounds__(256) void edgecnn_main_kernel(
    const float* __restrict__ x,      // [8192][13][512]
    const float* __restrict__ ws,     // Wfold (pair layout) + obias
    float* __restrict__ out) {        // [8192][13][13][4]
  __shared__ __align__(16) float ldsW[8192];        // folded B matrix, 32 KB
  __shared__ __align__(16) float ldsC[8 * 16 * 16]; // per-wave C scratch, 8 KB

  // Stage folded weights into LDS with the Tensor Data Mover: one 1-D tile of
  // 8192 f32 elements (32 KB), issued by wave 0 only (TDM ignores EXEC).
  if (threadIdx.x < 32) {
    unsigned long long ga = (unsigned long long)(uintptr_t)ws;
    unsigned int       la = (unsigned int)(uintptr_t)&ldsW[0]; // flat[31:0] == LDS byte addr

    u32x4 g0;
    g0[0] = 1u;                                  // count=1, user-mode load
    g0[1] = la;                                  // lds_addr (bytes)
    g0[2] = (unsigned int)ga;                    // global_addr[31:0]
    g0[3] = (unsigned int)((ga >> 32) & 0x1FFFFFFull) // global_addr[56:32]
          | 0x80000000u;                         // type = 2 ("image")

    i32x8 g1;
    g1[0] = 0x00020000;  // workgroup_mask=0, data_size=2 (4B), no flags
    g1[1] = 0x20000000;  // tensor_dim0 = 8192 (bits[79:48] -> dw1[31:16])
    g1[2] = 0x00010000;  // tensor_dim1 = 1    (bits[111:80] -> dw2[31:16])
    g1[3] = 0x20000000;  // tile_dim0  = 8192  (bits[127:112] -> dw3[31:16])
    g1[4] = 0;           // tile_dim1 = 0, tile_dim2 = 0 (1-D tile)
    g1[5] = 8192;        // tensor_dim0_stride[31:0]
    g1[6] = 0;
    g1[7] = 0;

    i32x4 gz = {0, 0, 0, 0};
#if defined(__clang_major__) && (__clang_major__ >= 23)
    i32x8 gz8 = {0, 0, 0, 0, 0, 0, 0, 0};
    __builtin_amdgcn_tensor_load_to_lds(g0, g1, gz, gz, gz8, 0);
#else
    __builtin_amdgcn_tensor_load_to_lds(g0, g1, gz, gz, 0);
#endif
    __builtin_amdgcn_s_wait_tensorcnt(0);
  }
  __syncthreads();

  const int wave = threadIdx.x >> 5;
  const int lane = threadIdx.x & 31;
  const int b    = blockIdx.x * 8 + wave;   // one batch element per wave
  const int n    = lane & 15;               // A: row M / B,C: col N
  const int h    = lane >> 4;               // lane-half selects K offset
  const int a    = (n > NATOMS - 1) ? (NATOMS - 1) : n;  // clamp padded rows

  // A operand: lane supplies x[b, a, k0 + 2h .. k0 + 2h + 1] each iteration.
  const float* ap = x + ((size_t)b * NATOMS + a) * DM + 2 * h;
  // B operand: pair (K = k0+2h, k0+2h+1) for column n, contiguous float2 in LDS.
  const v2f* bp = (const v2f*)ldsW + h * 16 + n;

  v8f c = {0.f, 0.f, 0.f, 0.f, 0.f, 0.f, 0.f, 0.f};
  #pragma unroll 8
  for (int kc = 0; kc < DM / 4; ++kc) {     // 128 x V_WMMA_F32_16X16X4_F32
    v2f av = *(const v2f*)(ap + 4 * kc);
    v2f bv = bp[32 * kc];
    c = __builtin_amdgcn_wmma_f32_16x16x4_f32(
        /*neg_a=*/false, av, /*neg_b=*/false, bv,
        /*c_mod=*/(short)0, c, /*reuse_a=*/false, /*reuse_b=*/false);
  }

  // Spill C tile: lane holds rows M = v + 8h, col N = n.
  float* cw = ldsC + wave * 256;
  #pragma unroll
  for (int v = 0; v < 8; ++v) cw[(v + 8 * h) * 16 + n] = c[v];
  __syncthreads();

  // Combine: out[b,i,j,o] = g_lo[min(i,j),o] + g_hi[max(i,j),o] + obias[o].
  const float ob = ws[8192 + (lane & 3)];   // o == idx & 3 is lane-invariant
  float* outb = out + (size_t)b * OUT_PER_B;
  for (int idx = lane; idx < OUT_PER_B; idx += 32) {
    int o = idx & 3;
    int p = idx >> 2;                  // i*13 + j
    int i = p / NATOMS;
    int j = p - i * NATOMS;
    int lo = i < j ? i : j;
    int hi = i < j ? j : i;
    outb[idx] = cw[lo * 16 + o] + cw[hi * 16 + 4 + o] + ob;
  }
}

// ----------------------------- launcher --------------------------------------
extern "C" void kernel_launch(void* const* d_in, const int* in_sizes, int n_in,
                              void* d_out, int out_size, void* d_ws, size_t ws_size,
                              hipStream_t stream) {
  const float* x  = (const float*)d_in[0];   // [8192,13,512]
  const float* W1 = (const float*)d_in[1];   // [100,512]
  const float* b1 = (const float*)d_in[2];   // [100]
  const float* Wc = (const float*)d_in[3];   // [4,200]
  const float* bc = (const float*)d_in[4];   // [4]
  float* out = (float*)d_out;
  float* ws  = (float*)d_ws;                 // needs 8196 floats (~33 KB)

  const int B = in_sizes[0] / (NATOMS * DM); // 8192

  edgecnn_fold_kernel<<<32, 256, 0, stream>>>(W1, b1, Wc, bc, ws);
  edgecnn_main_kernel<<<B / 8, 256, 0, stream>>>(x, ws, out);
}